// VposeFiled_Vjmlp_weight_52467320488330
// MI455X (gfx1250) — compile-verified
//
#include <hip/hip_runtime.h>

typedef __attribute__((ext_vector_type(2))) float v2f;
typedef __attribute__((ext_vector_type(8))) float v8f;

#define VJ       80
#define BATCH    2048
#define RFFK     240
#define BT       32          // batch rows per workgroup
#define NTHREADS 256         // 8 waves (wave32)
#define NWAVES   8
#define XS       484         // padded LDS stride for 480-wide input tile (even -> 8B aligned)
#define BUFA_FLOATS 15488    // 32 * 484
#define BUFB_FLOATS 16512    // 32 * 516 (layer-0 output, widest)
#define SMEM_FLOATS (BUFA_FLOATS + BUFB_FLOATS)   // 32000 floats = 128000 B

#define WMMA_F32(a, b, c) \
    __builtin_amdgcn_wmma_f32_16x16x4_f32(false, (a), false, (b), (short)0, (c), false, false)

// ---------------------------------------------------------------------------
// Kernel A: weight-norm row scales  s[k][o] = g[k][o] / ||v[k][o][:]||
// ---------------------------------------------------------------------------
__global__ void __launch_bounds__(256)
wnorm_scales_kernel(const float* __restrict__ v0, const float* __restrict__ g0,
                    const float* __restrict__ v1, const float* __restrict__ g1,
                    const float* __restrict__ v2, const float* __restrict__ g2,
                    const float* __restrict__ v3, const float* __restrict__ g3,
                    const float* __restrict__ v4, const float* __restrict__ g4,
                    float* __restrict__ scales)
{
    const int joint = blockIdx.x;
    const int tid   = threadIdx.x;
    const float* V[5] = {v0, v1, v2, v3, v4};
    const float* G[5] = {g0, g1, g2, g3, g4};
    const int CIN[5]   = {480, 512, 256, 128, 64};
    const int COUT[5]  = {512, 256, 128, 64, 6};
    const int WSOFF[5] = {0, 40960, 61440, 71680, 76800};
#pragma unroll
    for (int l = 0; l < 5; ++l) {
        const int cin = CIN[l], cout = COUT[l];
        const float* v = V[l] + (size_t)joint * cout * cin;
        const float* g = G[l] + (size_t)joint * cout;
        float* s = scales + WSOFF[l] + (size_t)joint * cout;
        for (int o = tid; o < cout; o += 256) {
            float sum = 0.f;
            const float* row = v + (size_t)o * cin;
            for (int i = 0; i < cin; ++i) { float x = row[i]; sum += x * x; }
            s[o] = g[o] / sqrtf(sum);
        }
    }
}

// ---------------------------------------------------------------------------
// Kernel B: fused RFF + 5 grouped weight-norm layers, one (joint, 32-row) tile
// per workgroup, intermediates in LDS, fp32 WMMA (16x16x4).
// Layers 0-1 (90% of FLOPs): 2Mx2N tiles per wave -> 1.0 mem-op per wmma.
// Layers 2-4: 2Mx1N tiles (small N), layer 4 streams straight to HBM.
// ---------------------------------------------------------------------------
__global__ void __launch_bounds__(NTHREADS)
vjmlp_fused_kernel(const float* __restrict__ qp,   const float* __restrict__ pf,
                   const float* __restrict__ adjm, const float* __restrict__ brff,
                   const float* __restrict__ v0, const float* __restrict__ v1,
                   const float* __restrict__ v2, const float* __restrict__ v3,
                   const float* __restrict__ v4,
                   const float* __restrict__ b0, const float* __restrict__ b1,
                   const float* __restrict__ b2, const float* __restrict__ b3,
                   const float* __restrict__ b4,
                   const float* __restrict__ scales,
                   float* __restrict__ out)
{
    extern __shared__ float smem[];
    const int joint = blockIdx.x;
    const int mbase = blockIdx.y * BT;
    const int tid   = threadIdx.x;
    const int lane  = tid & 31;
    const int wave  = tid >> 5;
    const int lhalf = lane >> 4;      // 0: lanes 0-15, 1: lanes 16-31
    const int lm    = lane & 15;

    float* bufA = smem;
    float* bufB = smem + BUFA_FLOATS;

    // ---- Stage 1: X0 = point_features + [cos(2pi qp@B), sin(2pi qp@B)] -> bufA
    for (int idx = tid; idx < BT * RFFK; idx += NTHREADS) {
        const int row = idx / RFFK;
        const int j   = idx - row * RFFK;
        const float* q = qp + ((size_t)(mbase + row) * VJ + joint) * 3;
        const float e = 6.28318530717958647692f *
            (q[0] * brff[j] + q[1] * brff[RFFK + j] + q[2] * brff[2 * RFFK + j]);
        float sv, cv;
        sincosf(e, &sv, &cv);
        const float* p = pf + ((size_t)(mbase + row) * VJ + joint) * 480;
        bufA[row * XS + j]        = cv + p[j];
        bufA[row * XS + RFFK + j] = sv + p[RFFK + j];
    }
    float adj = adjm[joint * VJ + joint];
    adj = adj > 0.f ? adj : 0.f;                // relu(diag(adjmat))[k]
    __syncthreads();

    const float* V[5] = {v0, v1, v2, v3, v4};
    const float* B[5] = {b0, b1, b2, b3, b4};
    const int CIN[5]   = {480, 512, 256, 128, 64};
    const int COUT[5]  = {512, 256, 128, 64, 6};
    const int WSOFF[5] = {0, 40960, 61440, 71680, 76800};

    float* cur = bufA;  int curStride = XS;     // current activations (LDS)
    float* nxt = bufB;                          // next activations (LDS)

#pragma unroll
    for (int l = 0; l < 5; ++l) {
        const int cin  = CIN[l];
        const int cout = COUT[l];
        const int outStride = cout + 4;         // even -> 8B-aligned rows
        const float* vbase = V[l] + (size_t)joint * cout * cin;
        const float* bbase = B[l] + (size_t)joint * cout;
        const float* sbase = scales + WSOFF[l] + (size_t)joint * cout;
        const int numN = (cout + 15) >> 4;

        // A operands (LDS), shared across all N tiles of this wave
        const float* a0 = cur + lm * curStride + 2 * lhalf;         // rows 0-15
        const float* a1 = cur + (16 + lm) * curStride + 2 * lhalf;  // rows 16-31

        if (cout >= 256) {
            // ---- 2M x 2N per wave: numN is a multiple of 16 here -------------
            for (int nt = wave; nt < numN; nt += 2 * NWAVES) {
                const int ncolA = (nt << 4) + lm;
                const int ncolB = ((nt + NWAVES) << 4) + lm;
                const float* bptrA = vbase + (size_t)ncolA * cin + 2 * lhalf;
                const float* bptrB = vbase + (size_t)ncolB * cin + 2 * lhalf;

                __builtin_prefetch(bptrA + ((size_t)2 * NWAVES << 4) * cin, 0, 1);

                v8f acc00 = {0.f,0.f,0.f,0.f,0.f,0.f,0.f,0.f};
                v8f acc10 = {0.f,0.f,0.f,0.f,0.f,0.f,0.f,0.f};
                v8f acc01 = {0.f,0.f,0.f,0.f,0.f,0.f,0.f,0.f};
                v8f acc11 = {0.f,0.f,0.f,0.f,0.f,0.f,0.f,0.f};

#pragma unroll 4
                for (int k = 0; k < cin; k += 4) {
                    const v2f bA  = *(const v2f*)(bptrA + k);   // global b64 (L2)
                    const v2f bBm = *(const v2f*)(bptrB + k);   // global b64 (L2)
                    const v2f am0 = *(const v2f*)(a0 + k);      // ds b64
                    const v2f am1 = *(const v2f*)(a1 + k);      // ds b64
                    acc00 = WMMA_F32(am0, bA,  acc00);
                    acc10 = WMMA_F32(am1, bA,  acc10);
                    acc01 = WMMA_F32(am0, bBm, acc01);
                    acc11 = WMMA_F32(am1, bBm, acc11);
                }

                const float snA = sbase[ncolA], bnA = bbase[ncolA];
                const float snB = sbase[ncolB], bnB = bbase[ncolB];
#pragma unroll
                for (int r = 0; r < 8; ++r) {
                    // C/D layout: VGPR r -> M=r (lanes<16) / r+8 (lanes>=16), N=lane&15
                    float y;
                    y = acc00[r] * snA + bnA; y = (y >= 0.f ? y : 0.01f * y) * adj;
                    nxt[(r + 8 * lhalf) * outStride + ncolA] = y;
                    y = acc10[r] * snA + bnA; y = (y >= 0.f ? y : 0.01f * y) * adj;
                    nxt[(16 + r + 8 * lhalf) * outStride + ncolA] = y;
                    y = acc01[r] * snB + bnB; y = (y >= 0.f ? y : 0.01f * y) * adj;
                    nxt[(r + 8 * lhalf) * outStride + ncolB] = y;
                    y = acc11[r] * snB + bnB; y = (y >= 0.f ? y : 0.01f * y) * adj;
                    nxt[(16 + r + 8 * lhalf) * outStride + ncolB] = y;
                }
            }
        } else {
            // ---- 2M x 1N per wave (layers 2-4) -------------------------------
            for (int nt = wave; nt < numN; nt += NWAVES) {
                const int ncol = (nt << 4) + lm;
                const int ncl  = ncol < cout ? ncol : cout - 1;   // layer-4 tail clamp
                const float* bptr = vbase + (size_t)ncl * cin + 2 * lhalf;

                v8f acc0 = {0.f,0.f,0.f,0.f,0.f,0.f,0.f,0.f};
                v8f acc1 = {0.f,0.f,0.f,0.f,0.f,0.f,0.f,0.f};

#pragma unroll 4
                for (int k = 0; k < cin; k += 4) {
                    const v2f bm  = *(const v2f*)(bptr + k);
                    const v2f am0 = *(const v2f*)(a0 + k);
                    const v2f am1 = *(const v2f*)(a1 + k);
                    acc0 = WMMA_F32(am0, bm, acc0);
                    acc1 = WMMA_F32(am1, bm, acc1);
                }

                const float sn = sbase[ncl];
                const float bn = bbase[ncl];
                if (l < 4) {
#pragma unroll
                    for (int r = 0; r < 8; ++r) {
                        float y = acc0[r] * sn + bn;
                        y = y >= 0.f ? y : 0.01f * y;
                        if (l < 3) y *= adj;
                        nxt[(r + 8 * lhalf) * outStride + ncol] = y;

                        float z = acc1[r] * sn + bn;
                        z = z >= 0.f ? z : 0.01f * z;
                        if (l < 3) z *= adj;
                        nxt[(16 + r + 8 * lhalf) * outStride + ncol] = z;
                    }
                } else if (ncol < cout) {   // layer 4: 6 valid outputs -> HBM
#pragma unroll
                    for (int r = 0; r < 8; ++r) {
                        out[((size_t)(mbase + r + 8 * lhalf) * VJ + joint) * 6 + ncol] =
                            acc0[r] * sn + bn;
                        out[((size_t)(mbase + 16 + r + 8 * lhalf) * VJ + joint) * 6 + ncol] =
                            acc1[r] * sn + bn;
                    }
                }
            }
        }
        __syncthreads();
        float* t = cur; cur = nxt; nxt = t;
        curStride = outStride;
    }
}

// ---------------------------------------------------------------------------
extern "C" void kernel_launch(void* const* d_in, const int* in_sizes, int n_in,
                              void* d_out, int out_size, void* d_ws, size_t ws_size,
                              hipStream_t stream) {
    const float* qp   = (const float*)d_in[0];
    const float* pf   = (const float*)d_in[1];
    const float* adjm = (const float*)d_in[2];
    const float* brff = (const float*)d_in[3];
    const float* v0 = (const float*)d_in[4];
    const float* g0 = (const float*)d_in[5];
    const float* b0 = (const float*)d_in[6];
    const float* v1 = (const float*)d_in[7];
    const float* g1 = (const float*)d_in[8];
    const float* b1 = (const float*)d_in[9];
    const float* v2 = (const float*)d_in[10];
    const float* g2 = (const float*)d_in[11];
    const float* b2 = (const float*)d_in[12];
    const float* v3 = (const float*)d_in[13];
    const float* g3 = (const float*)d_in[14];
    const float* b3 = (const float*)d_in[15];
    const float* v4 = (const float*)d_in[16];
    const float* g4 = (const float*)d_in[17];
    const float* b4 = (const float*)d_in[18];
    float* scales = (float*)d_ws;
    float* out    = (float*)d_out;

    // Pass 1: weight-norm scales (tiny; 80 blocks, 309 KB of scratch)
    wnorm_scales_kernel<<<dim3(VJ), dim3(256), 0, stream>>>(
        v0, g0, v1, g1, v2, g2, v3, g3, v4, g4, scales);

    // Pass 2: fused MLP, 128 KB dynamic LDS per workgroup (2 WGs / 320 KB WGP)
    const size_t shmem = (size_t)SMEM_FLOATS * sizeof(float);
    hipFuncSetAttribute((const void*)vjmlp_fused_kernel,
                        hipFuncAttributeMaxDynamicSharedMemorySize, (int)shmem);
    dim3 grid(VJ, BATCH / BT);
    vjmlp_fused_kernel<<<grid, dim3(NTHREADS), shmem, stream>>>(
        qp, pf, adjm, brff, v0, v1, v2, v3, v4, b0, b1, b2, b3, b4, scales, out);
}